// ArcDecoder_51951924412792
// MI455X (gfx1250) — compile-verified
//
#include <hip/hip_runtime.h>

typedef __attribute__((ext_vector_type(2))) float v2f;
typedef __attribute__((ext_vector_type(8))) float v8f;

#define NN 768
#define HH 128

// ---------------------------------------------------------------------------
// Stage 1: A[n][h] = b1[h] + sum_k z[n][k] * W1[h][k]        (half 0)
//          B[n][h] =         sum_k z[n][k] * W1[h][HH + k]   (half 1)
// One wave (32 lanes) per 16x16 output tile, chained V_WMMA_F32_16X16X4_F32
// over K=128 (32 WMMAs).
//
// ISA layouts (CDNA5 ISA):
//  A 16x4 f32 : VGPR r, lane L -> A[M = L&15][K = r + 2*(L>>4)]
//  B 4x16 f32 : VGPR r, lane L -> B[K = r + 2*(L>>4)][Ncol = L&15]
//  C/D 16x16  : VGPR r, lane L -> C[M = r + 8*(L>>4)][Ncol = L&15]
// ---------------------------------------------------------------------------
__global__ __launch_bounds__(32)
void arcdec_stage1_wmma(const float* __restrict__ z,
                        const float* __restrict__ W1,
                        const float* __restrict__ b1,
                        float* __restrict__ AB) {
  const int lane   = threadIdx.x;        // 0..31
  const int n0     = blockIdx.x * 16;
  const int h0     = blockIdx.y * 16;
  const int halfId = blockIdx.z;         // 0 -> A (W1[:, :H]), 1 -> B (W1[:, H:])
  const int m      = lane & 15;
  const int kk     = (lane >> 4) << 1;   // 0 or 2

  const float* zrow  = z  + (size_t)(n0 + m) * HH;
  const float* w1row = W1 + (size_t)(h0 + m) * (2 * HH) + halfId * HH;

  v8f c = {0.f, 0.f, 0.f, 0.f, 0.f, 0.f, 0.f, 0.f};
  #pragma unroll 4
  for (int k = 0; k < HH; k += 4) {
    v2f a, b;
    a.x = zrow[k + kk + 0];
    a.y = zrow[k + kk + 1];
    b.x = w1row[k + kk + 0];
    b.y = w1row[k + kk + 1];
    // 8 args: (neg_a, A, neg_b, B, c_mod, C, reuse_a, reuse_b)
    c = __builtin_amdgcn_wmma_f32_16x16x4_f32(false, a, false, b, (short)0, c,
                                              false, false);
  }

  const float bias = (halfId == 0) ? b1[h0 + m] : 0.0f;
  float* dst = AB + (size_t)halfId * NN * HH;
  const int rowBase = n0 + ((lane >> 4) << 3);   // +8 for lanes 16..31
  #pragma unroll
  for (int r = 0; r < 8; ++r) {
    dst[(size_t)(rowBase + r) * HH + (h0 + m)] = c[r] + bias;
  }
}

// ---------------------------------------------------------------------------
// Stage 2: out[p(i,j)] = b2 + sum_k relu(A[i][k] + B[j][k]) * w2[k]
// 32x32 pair tile per 256-thread block. Each thread: 4 i-rows x 1 j-col.
// A-tile reads are wave-uniform (LDS broadcast); B stored padded [32][132]
// so lane-j float4 reads are bank-conflict-free.
// ---------------------------------------------------------------------------
__global__ __launch_bounds__(256)
void arcdec_stage2_pairs(const float* __restrict__ AB,
                         const float* __restrict__ W2,
                         const float* __restrict__ b2,
                         float* __restrict__ out) {
  __shared__ float As[32 * 128];     // [i][k]
  __shared__ float Bs[32][132];      // [j][k], padded (132*4 B row stride, 16B aligned)
  __shared__ float w2s[128];

  const int t  = threadIdx.x;        // 0..255
  const int j0 = blockIdx.x * 32;
  const int i0 = blockIdx.y * 32;

  const float* Ag = AB;
  const float* Bg = AB + (size_t)NN * HH;

  const float4* Ag4 = (const float4*)(Ag + (size_t)i0 * HH);
  const float4* Bg4 = (const float4*)(Bg + (size_t)j0 * HH);
  float4* As4 = (float4*)As;

  #pragma unroll
  for (int q = 0; q < 4; ++q) {
    const int idx = t + q * 256;     // 0..1023 : 32 rows * 32 float4/row
    const int row = idx >> 5;
    const int c4  = idx & 31;
    As4[idx] = Ag4[idx];
    *(float4*)&Bs[row][c4 * 4] = Bg4[idx];
  }
  if (t < 32) {
    ((float4*)w2s)[t] = ((const float4*)W2)[t];
  }
  __syncthreads();

  const int j  = t & 31;
  const int ig = t >> 5;             // 0..7 -> rows ig*4 .. ig*4+3
  const float* abase = As + (ig * 4) * 128;

  float acc[4] = {0.f, 0.f, 0.f, 0.f};
  for (int k = 0; k < 128; k += 4) {
    const float4 bq = *(const float4*)&Bs[j][k];
    const float4 wq = *(const float4*)&w2s[k];
    #pragma unroll
    for (int ii = 0; ii < 4; ++ii) {
      const float4 aq = *(const float4*)(abase + ii * 128 + k);
      float h;
      h = fmaxf(aq.x + bq.x, 0.f); acc[ii] = fmaf(h, wq.x, acc[ii]);
      h = fmaxf(aq.y + bq.y, 0.f); acc[ii] = fmaf(h, wq.y, acc[ii]);
      h = fmaxf(aq.z + bq.z, 0.f); acc[ii] = fmaf(h, wq.z, acc[ii]);
      h = fmaxf(aq.w + bq.w, 0.f); acc[ii] = fmaf(h, wq.w, acc[ii]);
    }
  }

  const float bias2 = b2[0];
  const int jj = j0 + j;
  #pragma unroll
  for (int ii = 0; ii < 4; ++ii) {
    const int i = i0 + ig * 4 + ii;
    if (i != jj) {
      const size_t p = (size_t)i * (NN - 1) + (size_t)jj - (jj > i ? 1 : 0);
      out[p] = acc[ii] + bias2;
    }
  }
}

// ---------------------------------------------------------------------------
// inputs (setup_inputs order): z[768*128] f32, W1[128*256] f32, b1[128] f32,
//                              W2[1*128] f32, b2[1] f32
// output: 768*767 f32
// workspace: A (768*128 f32) then B (768*128 f32) = 768 KB
// ---------------------------------------------------------------------------
extern "C" void kernel_launch(void* const* d_in, const int* in_sizes, int n_in,
                              void* d_out, int out_size, void* d_ws, size_t ws_size,
                              hipStream_t stream) {
  (void)in_sizes; (void)n_in; (void)out_size; (void)ws_size;

  const float* z  = (const float*)d_in[0];
  const float* W1 = (const float*)d_in[1];
  const float* b1 = (const float*)d_in[2];
  const float* W2 = (const float*)d_in[3];
  const float* b2 = (const float*)d_in[4];
  float* out = (float*)d_out;
  float* AB  = (float*)d_ws;           // [2][768][128]

  dim3 g1(NN / 16, HH / 16, 2);        // 48 x 8 x 2 tiles, 1 wave each
  arcdec_stage1_wmma<<<g1, 32, 0, stream>>>(z, W1, b1, AB);

  dim3 g2(NN / 32, NN / 32);           // 24 x 24 pair tiles
  arcdec_stage2_pairs<<<g2, 256, 0, stream>>>(AB, W2, b2, out);
}